// MambaBlock_3865470566987
// MI455X (gfx1250) — compile-verified
//
#include <hip/hip_runtime.h>
#include <hip/hip_bf16.h>

typedef __attribute__((ext_vector_type(2))) float v2f;
typedef __attribute__((ext_vector_type(8))) float v8f;

#define B_SZ    4
#define L_SEQ   2048
#define D_INF   128
#define D_MODEL 256
#define D_INNER 512
#define S_ST    16
#define N_DBL   48      // R + 2S = 16 + 32
#define N_CHUNK 32
#define CHUNK_L 64
#define M_TOK   (B_SZ * L_SEQ)   // 8192

__device__ __forceinline__ float sigmoid_f(float v) {
    return 1.0f / (1.0f + __expf(-v));
}

__device__ __forceinline__ v8f wmma_f32(v2f a, v2f b, v8f c) {
    return __builtin_amdgcn_wmma_f32_16x16x4_f32(
        /*neg_a=*/false, a, /*neg_b=*/false, b,
        /*c_mod=*/(short)0, c, /*reuse_a=*/false, /*reuse_b=*/false);
}

// ---------------------------------------------------------------------------
// GEMM: C[M,N] = A'[M,K] @ W[N,K]^T (+ bias). A' row m maps to A row
// (b, L-1-l) when flip!=0 (backward direction), else row m.
// One wave computes a 16(M) x (NT*16)(N) strip with V_WMMA_F32_16X16X4_F32,
// reusing the A fragment across NT column tiles. NT is a compile-time
// constant so the K-loop is branch-free straight-line code. The loop is
// software-pipelined one 4-deep K-step ahead: next step's (NT+1) fragment
// loads are issued before the current step's WMMAs, so the per-WMMA wait is
// loadcnt<=NT+1 rather than 0 and L2 latency overlaps matrix math.
// Block = (32,4) = 4 waves covering 64 rows; grid = (M/64, N/(NT*16)).
//
// Fragment layouts (CDNA5 ISA):
//   A 16x4 f32 : lane holds M = lane&15, K = {2*(lane>>4), 2*(lane>>4)+1}
//   B 4x16 f32 : lane holds N = lane&15, K = same pair (mirrored)
//   C/D 16x16  : vgpr v, lane l -> row = v + 8*(l>>4), col = l&15
// ---------------------------------------------------------------------------
template <int NT>
__global__ void gemm_wmma_f32_k(const float* __restrict__ A,
                                const float* __restrict__ W,
                                const float* __restrict__ bias,
                                float* __restrict__ C,
                                int M, int N, int K, int L, int flip)
{
    const int lane  = threadIdx.x;          // 0..31
    const int wave  = threadIdx.y;          // 0..3
    const int tileM = blockIdx.x * 4 + wave;
    const int row0  = tileM * 16;
    const int colBase = blockIdx.y * (NT * 16);

    const int lmod  = lane & 15;
    const int khalf = (lane >> 4) << 1;     // K offset 0 or 2 within a 4-step

    // A row for this lane (apply sequence flip for the backward direction)
    int mrow = row0 + lmod;
    int arow = mrow;
    if (flip) {
        int bb = mrow / L;
        int ll = mrow - bb * L;
        arow = bb * L + (L - 1 - ll);
    }
    const float* Ap = A + (size_t)arow * K + khalf;

    v8f acc0 = {}, acc1 = {}, acc2 = {}, acc3 = {};
    const float* W0 = W + (size_t)(colBase +  0 + lmod) * K + khalf;
    const float* W1 = (NT > 1) ? W + (size_t)(colBase + 16 + lmod) * K + khalf : W0;
    const float* W2 = (NT > 2) ? W + (size_t)(colBase + 32 + lmod) * K + khalf : W0;
    const float* W3 = (NT > 3) ? W + (size_t)(colBase + 48 + lmod) * K + khalf : W0;

    // ---- software pipeline: prologue loads for kk = 0 ----
    v2f a  = *(const v2f*)(Ap);
    v2f b0 = *(const v2f*)(W0);
    v2f b1{}, b2{}, b3{};
    if constexpr (NT > 1) b1 = *(const v2f*)(W1);
    if constexpr (NT > 2) b2 = *(const v2f*)(W2);
    if constexpr (NT > 3) b3 = *(const v2f*)(W3);

#pragma unroll 2
    for (int kk = 4; kk < K; kk += 4) {
        // issue next step's loads first...
        v2f an  = *(const v2f*)(Ap + kk);
        v2f bn0 = *(const v2f*)(W0 + kk);
        v2f bn1{}, bn2{}, bn3{};
        if constexpr (NT > 1) bn1 = *(const v2f*)(W1 + kk);
        if constexpr (NT > 2) bn2 = *(const v2f*)(W2 + kk);
        if constexpr (NT > 3) bn3 = *(const v2f*)(W3 + kk);
        // ...then consume the current fragments while they fly
        acc0 = wmma_f32(a, b0, acc0);
        if constexpr (NT > 1) acc1 = wmma_f32(a, b1, acc1);
        if constexpr (NT > 2) acc2 = wmma_f32(a, b2, acc2);
        if constexpr (NT > 3) acc3 = wmma_f32(a, b3, acc3);
        a = an; b0 = bn0;
        if constexpr (NT > 1) b1 = bn1;
        if constexpr (NT > 2) b2 = bn2;
        if constexpr (NT > 3) b3 = bn3;
    }
    // ---- epilogue: last step ----
    acc0 = wmma_f32(a, b0, acc0);
    if constexpr (NT > 1) acc1 = wmma_f32(a, b1, acc1);
    if constexpr (NT > 2) acc2 = wmma_f32(a, b2, acc2);
    if constexpr (NT > 3) acc3 = wmma_f32(a, b3, acc3);

    const int rbase = row0 + ((lane >> 4) << 3);
    {
        int col = colBase + lmod;
        float badd = bias ? bias[col] : 0.0f;
#pragma unroll
        for (int v = 0; v < 8; ++v)
            C[(size_t)(rbase + v) * N + col] = acc0[v] + badd;
    }
    if constexpr (NT > 1) {
        int col = colBase + 16 + lmod;
        float badd = bias ? bias[col] : 0.0f;
#pragma unroll
        for (int v = 0; v < 8; ++v)
            C[(size_t)(rbase + v) * N + col] = acc1[v] + badd;
    }
    if constexpr (NT > 2) {
        int col = colBase + 32 + lmod;
        float badd = bias ? bias[col] : 0.0f;
#pragma unroll
        for (int v = 0; v < 8; ++v)
            C[(size_t)(rbase + v) * N + col] = acc2[v] + badd;
    }
    if constexpr (NT > 3) {
        int col = colBase + 48 + lmod;
        float badd = bias ? bias[col] : 0.0f;
#pragma unroll
        for (int v = 0; v < 8; ++v)
            C[(size_t)(rbase + v) * N + col] = acc3[v] + badd;
    }
}

// ---------------------------------------------------------------------------
// Causal depthwise conv (K=4) + SiLU, reading the x-half of xz.
// xz layout: [M_TOK, 2*D_INNER]; x part at [:, 0:512].
// ---------------------------------------------------------------------------
__global__ void conv_silu_kernel(const float* __restrict__ xz,
                                 const float* __restrict__ conv_W,
                                 const float* __restrict__ conv_b,
                                 float* __restrict__ xc)
{
    int idx = blockIdx.x * blockDim.x + threadIdx.x;
    if (idx >= M_TOK * D_INNER) return;
    int d = idx & (D_INNER - 1);
    int m = idx >> 9;                       // token index b*L + l
    int l = m & (L_SEQ - 1);
    int mbase = m - l;                      // start of this sequence
    float acc = conv_b[d];
#pragma unroll
    for (int k = 0; k < 4; ++k) {
        int ls = l + k - 3;
        float xv = (ls >= 0) ? xz[(size_t)(mbase + ls) * (2 * D_INNER) + d] : 0.0f;
        acc = fmaf(conv_W[d * 4 + k], xv, acc);
    }
    xc[(size_t)m * D_INNER + d] = acc * sigmoid_f(acc);
}

// ---------------------------------------------------------------------------
// dt = softplus(dbl[:, :16] @ dt_proj_W^T + dt_proj_b)   (K = 16, VALU)
// ---------------------------------------------------------------------------
__global__ void dt_proj_softplus_kernel(const float* __restrict__ dbl,
                                        const float* __restrict__ dtW,
                                        const float* __restrict__ dtb,
                                        float* __restrict__ dtf)
{
    int idx = blockIdx.x * blockDim.x + threadIdx.x;
    if (idx >= M_TOK * D_INNER) return;
    int d = idx & (D_INNER - 1);
    int m = idx >> 9;
    const float* r = dbl + (size_t)m * N_DBL;
    const float* w = dtW + d * 16;
    float acc = dtb[d];
#pragma unroll
    for (int i = 0; i < 16; ++i) acc = fmaf(r[i], w[i], acc);
    float sp = (acc > 20.0f) ? acc : log1pf(__expf(acc));
    dtf[(size_t)m * D_INNER + d] = sp;
}

// ---------------------------------------------------------------------------
// Chunked selective scan. Combine op: (a1,b1)∘(a2,b2) = (a1*a2, a2*b1 + b2).
// Phase 1: per (b,d,chunk) compute chunk summary (prod a, partial h).
// ---------------------------------------------------------------------------
__global__ void scan_phase1(const float* __restrict__ dtf,
                            const float* __restrict__ xc,
                            const float* __restrict__ dbl,
                            const float* __restrict__ A_log,
                            float* __restrict__ cA,
                            float* __restrict__ cB)
{
    int idx = blockIdx.x * blockDim.x + threadIdx.x;  // B*D_INNER*N_CHUNK
    if (idx >= B_SZ * D_INNER * N_CHUNK) return;
    int c = idx & (N_CHUNK - 1);
    int d = (idx >> 5) & (D_INNER - 1);
    int b = idx >> 14;

    float Aneg[S_ST];
#pragma unroll
    for (int s = 0; s < S_ST; ++s) Aneg[s] = -__expf(A_log[d * S_ST + s]);

    float h[S_ST], ap[S_ST];
#pragma unroll
    for (int s = 0; s < S_ST; ++s) { h[s] = 0.0f; ap[s] = 1.0f; }

    int l0 = c * CHUNK_L;
    for (int l = l0; l < l0 + CHUNK_L; ++l) {
        size_t m = (size_t)b * L_SEQ + l;
        float dt = dtf[m * D_INNER + d];
        float xv = xc[m * D_INNER + d];
        const float* Brow = dbl + m * N_DBL + 16;   // Bc at [16:32)
        float dx = dt * xv;
#pragma unroll
        for (int s = 0; s < S_ST; ++s) {
            float a  = __expf(dt * Aneg[s]);
            float bb = dx * Brow[s];
            h[s]  = fmaf(a, h[s], bb);
            ap[s] *= a;
        }
    }
    size_t base = (size_t)idx * S_ST;
#pragma unroll
    for (int s = 0; s < S_ST; ++s) { cA[base + s] = ap[s]; cB[base + s] = h[s]; }
}

// Phase 2: per (b,d), exclusive prefix over the 32 chunk summaries.
__global__ void scan_phase2(const float* __restrict__ cA,
                            const float* __restrict__ cB,
                            float* __restrict__ carry)
{
    int idx = blockIdx.x * blockDim.x + threadIdx.x;  // B*D_INNER
    if (idx >= B_SZ * D_INNER) return;
    float h[S_ST];
#pragma unroll
    for (int s = 0; s < S_ST; ++s) h[s] = 0.0f;
    size_t base = (size_t)idx * N_CHUNK * S_ST;
    for (int c = 0; c < N_CHUNK; ++c) {
        size_t o = base + (size_t)c * S_ST;
#pragma unroll
        for (int s = 0; s < S_ST; ++s) {
            carry[o + s] = h[s];                       // carry-in for chunk c
            h[s] = fmaf(cA[o + s], h[s], cB[o + s]);
        }
    }
}

// Phase 3: replay each chunk from its carry-in; fuse y = Σ h*C + D*xc and
// the silu(z) gate.
__global__ void scan_phase3(const float* __restrict__ dtf,
                            const float* __restrict__ xc,
                            const float* __restrict__ dbl,
                            const float* __restrict__ A_log,
                            const float* __restrict__ D_skip,
                            const float* __restrict__ xz,
                            const float* __restrict__ carry,
                            float* __restrict__ y)
{
    int idx = blockIdx.x * blockDim.x + threadIdx.x;
    if (idx >= B_SZ * D_INNER * N_CHUNK) return;
    int c = idx & (N_CHUNK - 1);
    int d = (idx >> 5) & (D_INNER - 1);
    int b = idx >> 14;

    float Aneg[S_ST];
#pragma unroll
    for (int s = 0; s < S_ST; ++s) Aneg[s] = -__expf(A_log[d * S_ST + s]);

    float h[S_ST];
    size_t cb = (size_t)idx * S_ST;
#pragma unroll
    for (int s = 0; s < S_ST; ++s) h[s] = carry[cb + s];

    float dsk = D_skip[d];
    int l0 = c * CHUNK_L;
    for (int l = l0; l < l0 + CHUNK_L; ++l) {
        size_t m = (size_t)b * L_SEQ + l;
        float dt = dtf[m * D_INNER + d];
        float xv = xc[m * D_INNER + d];
        const float* Brow = dbl + m * N_DBL + 16;   // Bc
        const float* Crow = dbl + m * N_DBL + 32;   // Cc
        float dx = dt * xv;
        float acc = 0.0f;
#pragma unroll
        for (int s = 0; s < S_ST; ++s) {
            float a = __expf(dt * Aneg[s]);
            h[s] = fmaf(a, h[s], dx * Brow[s]);
            acc  = fmaf(h[s], Crow[s], acc);
        }
        float yv = acc + dsk * xv;
        float z  = xz[m * (2 * D_INNER) + D_INNER + d];
        y[m * D_INNER + d] = yv * (z * sigmoid_f(z));
    }
}

// ---------------------------------------------------------------------------
// Residual + LayerNorm over D_MODEL=256; writes into the dir half of d_out.
// Block = 256 threads (8 waves), wave shuffles + LDS cross-wave reduction.
// ---------------------------------------------------------------------------
__global__ void ln_residual_kernel(const float* __restrict__ mo,
                                   const float* __restrict__ u,
                                   const float* __restrict__ gamma,
                                   const float* __restrict__ beta,
                                   float* __restrict__ out, int dirOff)
{
    __shared__ float ssum[8], ssq[8];
    __shared__ float smu, srs;
    int m = blockIdx.x;
    int n = threadIdx.x;
    size_t base = (size_t)m * D_MODEL;
    float r = mo[base + n] + u[base + n];

    float s = r, q = r * r;
#pragma unroll
    for (int o = 16; o > 0; o >>= 1) {
        s += __shfl_down(s, o, 32);
        q += __shfl_down(q, o, 32);
    }
    if ((n & 31) == 0) { ssum[n >> 5] = s; ssq[n >> 5] = q; }
    __syncthreads();
    if (n == 0) {
        float ts = 0.0f, tq = 0.0f;
#pragma unroll
        for (int i = 0; i < 8; ++i) { ts += ssum[i]; tq += ssq[i]; }
        float mu  = ts * (1.0f / D_MODEL);
        float var = tq * (1.0f / D_MODEL) - mu * mu;
        smu = mu;
        srs = rsqrtf(var + 1e-5f);
    }
    __syncthreads();
    out[(size_t)m * (2 * D_MODEL) + dirOff + n] =
        gamma[n] * (r - smu) * srs + beta[n];
}

// ---------------------------------------------------------------------------
extern "C" void kernel_launch(void* const* d_in, const int* in_sizes, int n_in,
                              void* d_out, int out_size, void* d_ws, size_t ws_size,
                              hipStream_t stream)
{
    const float* x          = (const float*)d_in[0];
    const float* proj_W     = (const float*)d_in[1];
    const float* proj_b     = (const float*)d_in[2];
    const float* in_proj_W  = (const float*)d_in[3];
    const float* conv_W     = (const float*)d_in[4];
    const float* conv_b     = (const float*)d_in[5];
    const float* x_proj_W   = (const float*)d_in[6];
    const float* dt_proj_W  = (const float*)d_in[7];
    const float* dt_proj_b  = (const float*)d_in[8];
    const float* A_log      = (const float*)d_in[9];
    const float* D_skip     = (const float*)d_in[10];
    const float* out_proj_W = (const float*)d_in[11];
    const float* ln_gamma   = (const float*)d_in[12];
    const float* ln_beta    = (const float*)d_in[13];
    float* out = (float*)d_out;

    // Workspace carve-up (floats). Total ~115 MB — fits entirely in the
    // 192 MB L2, so intermediates never round-trip to HBM.
    float* ws    = (float*)d_ws;
    float* u     = ws; ws += (size_t)M_TOK * D_MODEL;                    // 8 MB
    float* xz    = ws; ws += (size_t)M_TOK * 2 * D_INNER;                // 32 MB
    float* xc    = ws; ws += (size_t)M_TOK * D_INNER;                    // 16 MB
    float* dbl   = ws; ws += (size_t)M_TOK * N_DBL;                      // 1.5 MB
    float* dtf   = ws; ws += (size_t)M_TOK * D_INNER;                    // 16 MB
    size_t chunkElems = (size_t)B_SZ * D_INNER * N_CHUNK * S_ST;
    float* cA    = ws; ws += chunkElems;                                 // 4 MB
    float* cB    = ws; ws += chunkElems;                                 // 4 MB
    float* carry = ws; ws += chunkElems;                                 // 4 MB
    float* yv    = ws; ws += (size_t)M_TOK * D_INNER;                    // 16 MB
    float* mo    = ws; ws += (size_t)M_TOK * D_MODEL;                    // 8 MB

    dim3 gblk(32, 4);   // 4 waves per block, 64 output rows per block

    for (int dir = 0; dir < 2; ++dir) {
        // u = (flip? reverse(x) : x) @ proj_W^T + proj_b
        gemm_wmma_f32_k<4><<<dim3(M_TOK / 64, D_MODEL / 64), gblk, 0, stream>>>(
            x, proj_W, proj_b, u, M_TOK, D_MODEL, D_INF, L_SEQ, dir);

        // xz = u @ in_proj_W^T
        gemm_wmma_f32_k<4><<<dim3(M_TOK / 64, (2 * D_INNER) / 64), gblk, 0, stream>>>(
            u, in_proj_W, nullptr, xz, M_TOK, 2 * D_INNER, D_MODEL, L_SEQ, 0);

        // depthwise causal conv + SiLU
        conv_silu_kernel<<<(M_TOK * D_INNER) / 256, 256, 0, stream>>>(
            xz, conv_W, conv_b, xc);

        // dbl = xc @ x_proj_W^T  (N=48 -> exactly 3 column tiles)
        gemm_wmma_f32_k<3><<<dim3(M_TOK / 64, 1), gblk, 0, stream>>>(
            xc, x_proj_W, nullptr, dbl, M_TOK, N_DBL, D_INNER, L_SEQ, 0);

        // dt = softplus(dt_raw @ dt_proj_W^T + dt_proj_b)
        dt_proj_softplus_kernel<<<(M_TOK * D_INNER) / 256, 256, 0, stream>>>(
            dbl, dt_proj_W, dt_proj_b, dtf);

        // chunked associative scan
        scan_phase1<<<(B_SZ * D_INNER * N_CHUNK) / 256, 256, 0, stream>>>(
            dtf, xc, dbl, A_log, cA, cB);
        scan_phase2<<<(B_SZ * D_INNER) / 256, 256, 0, stream>>>(cA, cB, carry);
        scan_phase3<<<(B_SZ * D_INNER * N_CHUNK) / 256, 256, 0, stream>>>(
            dtf, xc, dbl, A_log, D_skip, xz, carry, yv);

        // mamba_out = y @ out_proj_W^T
        gemm_wmma_f32_k<4><<<dim3(M_TOK / 64, D_MODEL / 64), gblk, 0, stream>>>(
            yv, out_proj_W, nullptr, mo, M_TOK, D_MODEL, D_INNER, L_SEQ, 0);

        // residual + layernorm -> d_out[..., dir*256 : dir*256+256]
        ln_residual_kernel<<<M_TOK, D_MODEL, 0, stream>>>(
            mo, u, ln_gamma, ln_beta, out, dir * D_MODEL);
    }
}